// GaussianVoxelizer_76347338654043
// MI455X (gfx1250) — compile-verified
//
#include <hip/hip_runtime.h>
#include <hip/hip_bf16.h>

// GaussianVoxelizer for MI455X (gfx1250, wave32).
// d_out layout: [640000] density floats, then [640000*18] feature floats.

typedef __attribute__((ext_vector_type(2))) float v2f;
typedef __attribute__((ext_vector_type(8))) float v8f;

#define NGAUSS    8192
#define NVOX      640000      // 200*200*16
#define FEATS_OFF 640000
#define NFEAT     17          // NUM_CLASSES-1 real feature channels
#define WD        3200        // W*D = 200*16

__global__ __launch_bounds__(256) void gv_splat_kernel(
    const float* __restrict__ means,   // [N,3]
    const float* __restrict__ opac,    // [N,1]
    const float* __restrict__ feats,   // [N,17]
    const float* __restrict__ scales,  // [N,3]
    const float* __restrict__ rots,    // [N,4] wxyz
    float* __restrict__ out)
{
    const int g    = blockIdx.x;
    const int lane = threadIdx.x & 31;
    const int wave = threadIdx.x >> 5;

    // ---- per-gaussian setup (redundant per thread; ~200 FLOPs, negligible) ----
    float qw = rots[g*4+0], qx = rots[g*4+1], qy = rots[g*4+2], qz = rots[g*4+3];
    float qn = rsqrtf(qw*qw + qx*qx + qy*qy + qz*qz);
    qw *= qn; qx *= qn; qy *= qn; qz *= qn;
    float R00 = 1.f-2.f*(qy*qy+qz*qz), R01 = 2.f*(qx*qy-qw*qz), R02 = 2.f*(qx*qz+qw*qy);
    float R10 = 2.f*(qx*qy+qw*qz),     R11 = 1.f-2.f*(qx*qx+qz*qz), R12 = 2.f*(qy*qz-qw*qx);
    float R20 = 2.f*(qx*qz-qw*qy),     R21 = 2.f*(qy*qz+qw*qx),     R22 = 1.f-2.f*(qx*qx+qy*qy);

    float s0 = scales[g*3+0], s1 = scales[g*3+1], s2 = scales[g*3+2];
    float t0 = s0*s0, t1 = s1*s1, t2 = s2*s2;
    float i0 = 1.f/t0, i1 = 1.f/t1, i2 = 1.f/t2;
    // Sigma^-1 = R diag(1/s^2) R^T  (symmetric)
    float P00 = R00*R00*i0 + R01*R01*i1 + R02*R02*i2;
    float P11 = R10*R10*i0 + R11*R11*i1 + R12*R12*i2;
    float P22 = R20*R20*i0 + R21*R21*i1 + R22*R22*i2;
    float P01 = R00*R10*i0 + R01*R11*i1 + R02*R12*i2;
    float P02 = R00*R20*i0 + R01*R21*i1 + R02*R22*i2;
    float P12 = R10*R20*i0 + R11*R21*i1 + R12*R22*i2;
    // sigma = sqrt(diag(Sigma)), Sigma = R diag(s^2) R^T
    float sg0 = sqrtf(R00*R00*t0 + R01*R01*t1 + R02*R02*t2);
    float sg1 = sqrtf(R10*R10*t0 + R11*R11*t1 + R12*R12*t2);
    float sg2 = sqrtf(R20*R20*t0 + R21*R21*t1 + R22*R22*t2);

    float mx = means[g*3+0], my = means[g*3+1], mz = means[g*3+2];
    float op = opac[g];

    const float LOx = -40.f, LOy = -40.f, LOz = -1.f;
    const float HIx =  40.f, HIy =  40.f, HIz =  5.4f;

    float blx = fminf(fmaxf(mx - 3.f*sg0, LOx), HIx), bhx = fminf(fmaxf(mx + 3.f*sg0, LOx), HIx);
    float bly = fminf(fmaxf(my - 3.f*sg1, LOy), HIy), bhy = fminf(fmaxf(my + 3.f*sg1, LOy), HIy);
    float blz = fminf(fmaxf(mz - 3.f*sg2, LOz), HIz), bhz = fminf(fmaxf(mz + 3.f*sg2, LOz), HIz);
    bool vx = ((blx > LOx) || (bhx > LOx)) && ((blx < HIx) || (bhx < HIx));
    bool vy = ((bly > LOy) || (bhy > LOy)) && ((bly < HIy) || (bhy < HIy));
    bool vz = ((blz > LOz) || (bhz > LOz)) && ((blz < HIz) || (bhz < HIz));
    bool valid = vx && vy && vz;
    if (!valid) return;   // uniform across block

    int ilx = (int)((blx - LOx) / 0.4f), ihx = (int)((bhx - LOx) / 0.4f);
    int ily = (int)((bly - LOy) / 0.4f), ihy = (int)((bhy - LOy) / 0.4f);
    int ilz = (int)((blz - LOz) / 0.4f), ihz = (int)((bhz - LOz) / 0.4f);

    // Compacted span box (inclusive [i_lo, i_hi]); clamp to static K=10 tile so
    // semantics match the reference's masked KxKxK tile exactly.
    int sx = min(ihx - ilx + 1, 10);
    int sy = min(ihy - ily + 1, 10);
    int sz = min(ihz - ilz + 1, 10);
    int syz = sy * sz;
    int vol = sx * syz;                 // 1..1000
    int nchunks = (vol + 31) >> 5;      // avg ~4 instead of fixed 32

    // B matrix for WMMA outer product: this gaussian's 16-class feature row,
    // replicated across both lane halves (robust to either 4x16 layout half);
    // the corresponding A K-slots are zero, so the product is exact.
    float featB = feats[g*NFEAT + (lane & 15)];
    float f16c  = feats[g*NFEAT + 16];
    v2f Bm; Bm[0] = featB; Bm[1] = 0.f;

    float* __restrict__ dOut = out;
    float* __restrict__ fOut = out + FEATS_OFF;

    for (int chunk = wave; chunk < nchunks; chunk += 8) {
        int idx = (chunk << 5) + lane;
        unsigned uidx = (unsigned)idx;
        int oi  = (int)(uidx / (unsigned)syz);
        int rem = idx - oi * syz;
        int oj  = (int)((unsigned)rem / (unsigned)sz);
        int ok  = rem - oj * sz;
        int vi = ilx + oi, vj = ily + oj, vk = ilz + ok;

        // oi/oj/ok are in-span by construction; only tail + grid-dim checks remain
        bool m = (idx < vol) && (vi < 200) && (vj < 200) && (vk < 16);

        float cx = ((float)vi + 0.5f) * 0.4f + LOx;
        float cy = ((float)vj + 0.5f) * 0.4f + LOy;
        float cz = ((float)vk + 0.5f) * 0.4f + LOz;
        float dx = cx - mx, dy = cy - my, dz = cz - mz;
        float maha = P00*dx*dx + P11*dy*dy + P22*dz*dz
                   + 2.f*(P01*dx*dy + P02*dx*dz + P12*dy*dz);
        float dens = m ? op * __expf(-0.5f * maha) : 0.f;
        int   flat = m ? (vi * WD + vj * 16 + vk) : -1;

        // density + class 16 (scalar channel); class 17 is zero for real gaussians
        if (flat >= 0) {
            atomicAdd(&dOut[flat], dens);
            atomicAdd(&fOut[flat * 18 + 16], dens * f16c);
        }

        // ---- WMMA outer products: D[v, c] = dens[v] * feat[c], classes 0..15 ----
        // A 16x4 f32 layout: VGPR0 = K0 (lanes 0-15) / K2 (lanes 16-31); VGPR1 = K1/K3.
        // dens goes in K0 for voxels 0..15; all other K slots zero. EXEC is full here
        // (all control flow above is wave-uniform or select-based).
        v2f A0; A0[0] = (lane < 16) ? dens : 0.f; A0[1] = 0.f;
        v8f D0 = {};
        D0 = __builtin_amdgcn_wmma_f32_16x16x4_f32(
                 false, A0, false, Bm, (short)0, D0, false, false);

        // second 16-voxel group: pull dens from the upper half-wave
        float densHi = __shfl(dens, lane ^ 16, 32);
        v2f A1; A1[0] = (lane < 16) ? densHi : 0.f; A1[1] = 0.f;
        v8f D1 = {};
        D1 = __builtin_amdgcn_wmma_f32_16x16x4_f32(
                 false, A1, false, Bm, (short)0, D1, false, false);

        // Scatter: D VGPR r, lanes 0-15 -> (M=r, N=lane); lanes 16-31 -> (M=r+8, N=lane-16)
        int rowBase = (lane >> 4) << 3;   // 0 or 8
        int cls     = lane & 15;
        #pragma unroll
        for (int r = 0; r < 8; ++r) {
            int fA = __shfl(flat, rowBase + r, 32);        // voxel M of group 0
            if (fA >= 0) atomicAdd(&fOut[fA * 18 + cls], D0[r]);
            int fB = __shfl(flat, 16 + rowBase + r, 32);   // voxel M of group 1
            if (fB >= 0) atomicAdd(&fOut[fB * 18 + cls], D1[r]);
        }
    }
}

__global__ __launch_bounds__(256) void gv_finalize_kernel(
    float* __restrict__ out, const float* __restrict__ empty_scalar)
{
    int v = blockIdx.x * 256 + threadIdx.x;
    if (v >= NVOX) return;
    int i = v / WD;
    int rem = v - i * WD;
    int j = rem >> 4;
    int k = rem & 15;

    // empty gaussian: mean (0,0,2.2), inv cov diag = 1/range^2, range = (80,80,6.4)
    float cx = ((float)i + 0.5f) * 0.4f - 40.f;
    float cy = ((float)j + 0.5f) * 0.4f - 40.f;
    float cz = ((float)k + 0.5f) * 0.4f - 1.f;
    float dzc = cz - 2.2f;
    float maha_e = cx*cx * (1.f/6400.f) + cy*cy * (1.f/6400.f) + dzc*dzc * (1.f/40.96f);
    float dens_e = __expf(-0.5f * maha_e);

    float d = out[v] + dens_e;
    out[v] = d;
    float invd = 1.f / fmaxf(d, 1e-6f);
    float es = empty_scalar[0];

    float* f = out + FEATS_OFF + v * 18;
    #pragma unroll
    for (int c = 0; c < 17; ++c) f[c] = f[c] * invd;
    f[17] = (f[17] + dens_e * es) * invd;
}

extern "C" void kernel_launch(void* const* d_in, const int* in_sizes, int n_in,
                              void* d_out, int out_size, void* d_ws, size_t ws_size,
                              hipStream_t stream) {
    const float* means  = (const float*)d_in[0];
    const float* opac   = (const float*)d_in[1];
    const float* feats  = (const float*)d_in[2];
    const float* scales = (const float*)d_in[3];
    const float* rots   = (const float*)d_in[4];
    const float* es     = (const float*)d_in[5];
    float* out = (float*)d_out;

    // zero accumulators (graph-capture safe)
    hipMemsetAsync(out, 0, (size_t)out_size * sizeof(float), stream);

    gv_splat_kernel<<<NGAUSS, 256, 0, stream>>>(means, opac, feats, scales, rots, out);
    gv_finalize_kernel<<<(NVOX + 255) / 256, 256, 0, stream>>>(out, es);
}